// HPUAttentionImpl_80384607912383
// MI455X (gfx1250) — compile-verified
//
#include <hip/hip_runtime.h>
#include <hip/hip_bf16.h>

typedef float v2f __attribute__((ext_vector_type(2)));
typedef float v8f __attribute__((ext_vector_type(8)));

#define NB   448
#define BPS  14
#define BATCH 32
#define NH   32
#define KVH  8
#define GQ   4
#define HD   128
#define BS   128
#define SCALE 0.08838834764831845f
#define WS_STRIDE (GQ*HD + 2*GQ)   // 520 floats per (block, kvh)

// Stage 256 contiguous bytes global -> LDS via the CDNA5 async LDS DMA.
// Per ISA 15.18.3 the offset immediate is added to both the LDS and the global address,
// so one offset immediate advances both sides.
__device__ __forceinline__ void stage256_async(unsigned lds_off, const float* src) {
    asm volatile(
        "global_load_async_to_lds_b128 %0, %1, off\n\t"
        "global_load_async_to_lds_b128 %0, %1, off offset:16\n\t"
        "global_load_async_to_lds_b128 %0, %1, off offset:32\n\t"
        "global_load_async_to_lds_b128 %0, %1, off offset:48\n\t"
        "global_load_async_to_lds_b128 %0, %1, off offset:64\n\t"
        "global_load_async_to_lds_b128 %0, %1, off offset:80\n\t"
        "global_load_async_to_lds_b128 %0, %1, off offset:96\n\t"
        "global_load_async_to_lds_b128 %0, %1, off offset:112\n\t"
        "global_load_async_to_lds_b128 %0, %1, off offset:128\n\t"
        "global_load_async_to_lds_b128 %0, %1, off offset:144\n\t"
        "global_load_async_to_lds_b128 %0, %1, off offset:160\n\t"
        "global_load_async_to_lds_b128 %0, %1, off offset:176\n\t"
        "global_load_async_to_lds_b128 %0, %1, off offset:192\n\t"
        "global_load_async_to_lds_b128 %0, %1, off offset:208\n\t"
        "global_load_async_to_lds_b128 %0, %1, off offset:224\n\t"
        "global_load_async_to_lds_b128 %0, %1, off offset:240\n\t"
        :: "v"(lds_off), "v"(src) : "memory");
}

__device__ __forceinline__ void wait_async_zero() {
    asm volatile("s_wait_asynccnt 0x0" ::: "memory");
}

// ---------------------------------------------------------------------------
// Pass 1: per (kv-block n, kv-head) flash-decoding partial
// ---------------------------------------------------------------------------
__global__ __launch_bounds__(128) void pa_block_kernel(
    const float* __restrict__ query,
    const float* __restrict__ key_new,
    const float* __restrict__ value_new,
    const float* __restrict__ key_cache,
    const float* __restrict__ value_cache,
    const int*   __restrict__ block_list,
    const int*   __restrict__ block_groups,
    const float* __restrict__ block_bias,
    const int*   __restrict__ block_indices,
    const int*   __restrict__ block_offsets,
    const float* __restrict__ alibi_blocks,
    const float* __restrict__ alibi_slopes,
    float*       __restrict__ ws)
{
    __shared__ float tile[BS * 64];      // K half [128 slots][64 d]  OR  V half [64 slots][128 d]
    __shared__ float qbuf[GQ * HD];
    __shared__ float scores[GQ * BS];
    __shared__ float red[GQ * 32];
    __shared__ float opart[4 * GQ * HD];
    __shared__ float mbuf[GQ], lbuf[GQ];

    const int n    = blockIdx.x / KVH;
    const int kvh  = blockIdx.x % KVH;
    const int t    = threadIdx.x;
    const int lane = t & 31;
    const int wave = t >> 5;

    const int b        = block_groups[n];
    const int cblk     = block_list[n];
    const bool is_last = (cblk == block_indices[b]);
    const int new_off  = block_offsets[b];

    const float* kbase = key_cache   + (size_t)cblk * BS * KVH * HD + (size_t)kvh * HD;
    const float* vbase = value_cache + (size_t)cblk * BS * KVH * HD + (size_t)kvh * HD;
    const float* knew  = key_new     + (size_t)b * KVH * HD + (size_t)kvh * HD;
    const float* vnew  = value_new   + (size_t)b * KVH * HD + (size_t)kvh * HD;

    // scaled query rows for this kv head
    for (int i = t; i < GQ * HD; i += 128) {
        int g = i >> 7, d = i & (HD - 1);
        qbuf[i] = SCALE * query[(size_t)b * NH * HD + (size_t)(kvh * GQ + g) * HD + d];
    }

    // warm the V stream while we chew on K (global_prefetch_b8)
    __builtin_prefetch((const void*)(vbase + (size_t)t * KVH * HD), 0, 0);

    const int m16  = lane & 15;
    const int koff = (lane >> 4) << 1;   // 0 for lanes 0-15, 2 for lanes 16-31
    const int st0  = wave * 2;           // this wave's two 16-slot s-tiles

    // ---------------- QK^T via v_wmma_f32_16x16x4_f32 ----------------
    v8f c0 = {}, c1 = {};
    for (int half = 0; half < 2; ++half) {
        __syncthreads();
        const int dh = half * 64;
        {   // async-stage K[:, dh:dh+64] into LDS, one slot per thread (256B each)
            const int s = t;
            const float* src = (is_last && s == new_off) ? (knew + dh)
                                                         : (kbase + (size_t)s * KVH * HD + dh);
            stage256_async((unsigned)(size_t)(tile + s * 64), src);
        }
        wait_async_zero();
        __syncthreads();
        #pragma unroll
        for (int ks = 0; ks < 16; ++ks) {
            const int dl = ks * 4 + koff;
            v2f a;
            if (m16 < GQ) { a.x = qbuf[m16 * HD + dh + dl]; a.y = qbuf[m16 * HD + dh + dl + 1]; }
            else          { a.x = 0.0f;                     a.y = 0.0f; }
            v2f b0, b1;
            b0.x = tile[(st0 * 16 + m16) * 64 + dl];
            b0.y = tile[(st0 * 16 + m16) * 64 + dl + 1];
            b1.x = tile[((st0 + 1) * 16 + m16) * 64 + dl];
            b1.y = tile[((st0 + 1) * 16 + m16) * 64 + dl + 1];
            c0 = __builtin_amdgcn_wmma_f32_16x16x4_f32(false, a, false, b0, (short)0, c0, false, false);
            c1 = __builtin_amdgcn_wmma_f32_16x16x4_f32(false, a, false, b1, (short)0, c1, false, false);
        }
    }

    // ---------------- bias + score writeback ----------------
    if (lane < 16) {
        #pragma unroll
        for (int r = 0; r < GQ; ++r) {
            const float sl = alibi_slopes[kvh * GQ + r];
            const int s0 = st0 * 16 + lane;
            const int s1 = (st0 + 1) * 16 + lane;
            scores[r * BS + s0] = c0[r] + alibi_blocks[n * BS + s0] * sl + block_bias[n * BS + s0];
            scores[r * BS + s1] = c1[r] + alibi_blocks[n * BS + s1] * sl + block_bias[n * BS + s1];
        }
    }
    __syncthreads();

    // ---------------- block-local softmax stats ----------------
    {
        const int g = t >> 5, j = t & 31;
        float vm = scores[g * BS + j];
        vm = fmaxf(vm, scores[g * BS + j + 32]);
        vm = fmaxf(vm, scores[g * BS + j + 64]);
        vm = fmaxf(vm, scores[g * BS + j + 96]);
        red[g * 32 + j] = vm;
    }
    __syncthreads();
    if (t < GQ) {
        float vm = red[t * 32];
        for (int j = 1; j < 32; ++j) vm = fmaxf(vm, red[t * 32 + j]);
        mbuf[t] = vm;
    }
    __syncthreads();
    {
        const int g = t >> 5, j = t & 31;
        const float m = mbuf[g];
        float ssum = 0.0f;
        #pragma unroll
        for (int q = 0; q < 4; ++q) {
            const int s = j + q * 32;
            const float p = expf(scores[g * BS + s] - m);
            scores[g * BS + s] = p;
            ssum += p;
        }
        red[g * 32 + j] = ssum;
    }
    __syncthreads();
    if (t < GQ) {
        float sm = 0.0f;
        for (int j = 0; j < 32; ++j) sm += red[t * 32 + j];
        lbuf[t] = sm;
    }

    // ---------------- PV via v_wmma_f32_16x16x4_f32 ----------------
    v8f o[8] = {};
    for (int half = 0; half < 2; ++half) {
        __syncthreads();
        {   // async-stage V[half*64 : half*64+64, :] into LDS (two threads/slot)
            const int s   = half * 64 + (t >> 1);
            const int dh2 = (t & 1) * 64;
            const float* src = (is_last && s == new_off) ? (vnew + dh2)
                                                         : (vbase + (size_t)s * KVH * HD + dh2);
            stage256_async((unsigned)(size_t)(tile + (t >> 1) * HD + dh2), src);
        }
        wait_async_zero();
        __syncthreads();
        #pragma unroll
        for (int ks = 0; ks < 4; ++ks) {
            const int sl = wave * 16 + ks * 4 + koff;   // local slot (K row) for this lane
            const int sg = half * 64 + sl;              // global slot for p lookup
            v2f a;
            if (m16 < GQ) { a.x = scores[m16 * BS + sg]; a.y = scores[m16 * BS + sg + 1]; }
            else          { a.x = 0.0f;                  a.y = 0.0f; }
            #pragma unroll
            for (int dt = 0; dt < 8; ++dt) {
                v2f bm;
                bm.x = tile[sl * HD + dt * 16 + m16];
                bm.y = tile[(sl + 1) * HD + dt * 16 + m16];
                o[dt] = __builtin_amdgcn_wmma_f32_16x16x4_f32(false, a, false, bm, (short)0, o[dt], false, false);
            }
        }
    }

    // ---------------- cross-wave reduce + store partial ----------------
    __syncthreads();
    if (lane < 16) {
        #pragma unroll
        for (int r = 0; r < GQ; ++r)
            #pragma unroll
            for (int dt = 0; dt < 8; ++dt)
                opart[(wave * GQ + r) * HD + dt * 16 + lane] = o[dt][r];
    }
    __syncthreads();

    float* wsp = ws + (size_t)blockIdx.x * WS_STRIDE;
    for (int i = t; i < GQ * HD; i += 128) {
        wsp[i] = opart[i] + opart[GQ * HD + i] + opart[2 * GQ * HD + i] + opart[3 * GQ * HD + i];
    }
    if (t < GQ) {
        wsp[GQ * HD + t]      = mbuf[t];
        wsp[GQ * HD + GQ + t] = lbuf[t];
    }
}

// ---------------------------------------------------------------------------
// Pass 2: combine the 14 block partials per (batch, kv-head)
// ---------------------------------------------------------------------------
__global__ __launch_bounds__(128) void pa_combine_kernel(
    const float* __restrict__ ws, float* __restrict__ out)
{
    const int b   = blockIdx.x / KVH;
    const int kvh = blockIdx.x % KVH;
    const int t   = threadIdx.x;
    const int g   = t >> 5;
    const int j   = t & 31;

    float M = -3.402823466e+38f;
    for (int i = 0; i < BPS; ++i) {
        const float* wsp = ws + (size_t)((b * BPS + i) * KVH + kvh) * WS_STRIDE;
        M = fmaxf(M, wsp[GQ * HD + g]);
    }
    float L = 0.0f;
    float acc[4] = {0.0f, 0.0f, 0.0f, 0.0f};
    for (int i = 0; i < BPS; ++i) {
        const float* wsp = ws + (size_t)((b * BPS + i) * KVH + kvh) * WS_STRIDE;
        const float w = expf(wsp[GQ * HD + g] - M);
        L += w * wsp[GQ * HD + GQ + g];
        #pragma unroll
        for (int q = 0; q < 4; ++q)
            acc[q] += w * wsp[g * HD + j + q * 32];
    }
    const float inv = 1.0f / L;
    #pragma unroll
    for (int q = 0; q < 4; ++q)
        out[(size_t)b * NH * HD + (size_t)(kvh * GQ + g) * HD + j + q * 32] = acc[q] * inv;
}

// ---------------------------------------------------------------------------
extern "C" void kernel_launch(void* const* d_in, const int* in_sizes, int n_in,
                              void* d_out, int out_size, void* d_ws, size_t ws_size,
                              hipStream_t stream) {
    const float* query       = (const float*)d_in[0];
    const float* key_new     = (const float*)d_in[1];
    const float* value_new   = (const float*)d_in[2];
    const float* key_cache   = (const float*)d_in[3];
    const float* value_cache = (const float*)d_in[4];
    const int*   block_list  = (const int*)d_in[5];
    const int*   block_grp   = (const int*)d_in[6];
    // d_in[7] block_mapping: unused (block_groups carries the same info)
    const float* block_bias  = (const float*)d_in[8];
    const int*   block_idx   = (const int*)d_in[9];
    const int*   block_off   = (const int*)d_in[10];
    const float* alibi_blk   = (const float*)d_in[11];
    const float* alibi_slp   = (const float*)d_in[12];
    float* ws  = (float*)d_ws;
    float* out = (float*)d_out;

    pa_block_kernel<<<NB * KVH, 128, 0, stream>>>(
        query, key_new, value_new, key_cache, value_cache,
        block_list, block_grp, block_bias, block_idx, block_off,
        alibi_blk, alibi_slp, ws);
    pa_combine_kernel<<<BATCH * KVH, 128, 0, stream>>>(ws, out);
}